// Attention_17119739642103
// MI455X (gfx1250) — compile-verified
//
#include <hip/hip_runtime.h>
#include <math.h>

// ---------------------------------------------------------------------------
// Problem constants (B,N,D,O fixed by the reference)
// ---------------------------------------------------------------------------
#define B_ 16
#define N_ 16384
#define D_ 256
#define O_ 256
#define SCALE_ 0.0625f          // 1/sqrt(O)

#define SLICES1 64
#define ROWS1   (N_ / SLICES1)  // 256 rows per block in pass 1
#define ROWS3   512             // rows per block in fused score/agg pass
#define NB3     (N_ / ROWS3)    // 32 blocks per batch

// Workspace layout (float offsets)
#define OFF_P1   0u              // [B][SLICES1][D]  = 262144
#define OFF_V    262144u         // [B][D]           = 4096
#define OFF_C    266240u         // [B]              = 16
#define OFF_S    266256u         // [B][N]           = 262144
#define OFF_PMAX 528400u         // [B][NB3]         = 512
#define OFF_PSUM 528912u         // [B][NB3]         = 512
#define OFF_M    529424u         // [B]
#define OFF_LINV 529440u         // [B]
#define OFF_AGGP 529456u         // [B][NB3][D]      = 131072
// total: 660528 floats ~= 2.52 MB of d_ws

typedef __attribute__((ext_vector_type(2))) float v2f;
typedef __attribute__((ext_vector_type(8))) float v8f;

// ---------------------------------------------------------------------------
// Pass 1: partial column sums of x   partial[b][slice][d] = sum_rows x[b,n,d]
// ---------------------------------------------------------------------------
__global__ __launch_bounds__(256)
void k1_partial_sum(const float* __restrict__ x, float* __restrict__ ws) {
    int blk = blockIdx.x;                // b*SLICES1 + s
    int b = blk / SLICES1, s = blk % SLICES1;
    int d = threadIdx.x;                 // 256 threads == D
    const float* xp = x + ((size_t)(b * N_ + s * ROWS1)) * D_ + d;
    float acc = 0.f;
    for (int r = 0; r < ROWS1; ++r) {
        if (r + 16 < ROWS1) __builtin_prefetch(xp + (size_t)(r + 16) * D_, 0, 0);
        acc += xp[(size_t)r * D_];
    }
    ws[OFF_P1 + (unsigned)((b * SLICES1 + s) * D_ + d)] = acc;
}

// ---------------------------------------------------------------------------
// Pass 2: single workgroup, 512 threads = 16 full waves (EXEC all ones).
//   A) xsum[b][d]    = reduce partials           (into LDS)
//   B) ksum[b][o]    = xsum . Wk^T + N*bk        (WMMA f32 16x16x4)
//   C) v[b][d]       = scale * (ksum . Wq)       (WMMA f32 16x16x4)
//      c[b]          = scale * (bq . ksum[b])
// ---------------------------------------------------------------------------
__global__ __launch_bounds__(512)
void k2_small_gemms(const float* __restrict__ Wq, const float* __restrict__ bq,
                    const float* __restrict__ Wk, const float* __restrict__ bk,
                    float* __restrict__ ws) {
    __shared__ float xsum[B_][D_];
    __shared__ float ksum[B_][O_];
    int tid = threadIdx.x;

    // --- stage A: reduce pass-1 partials into LDS ---
    for (int i = tid; i < B_ * D_; i += 512) {
        int b = i >> 8, d = i & 255;
        const float* p = ws + OFF_P1 + (unsigned)((b * SLICES1) * D_ + d);
        float s = 0.f;
        for (int k = 0; k < SLICES1; ++k) s += p[(size_t)k * D_];
        xsum[b][d] = s;
    }
    __syncthreads();

    int lane = tid & 31;
    int wv   = tid >> 5;        // 0..15, one 16x16 output tile per wave
    int half = lane >> 4;       // 0: lanes 0-15 (K 0,1)  1: lanes 16-31 (K 2,3)
    int l16  = lane & 15;       // M index for A, N index for B/C/D

    // --- stage B: ksum = xsum * Wk^T + N*bk ---
    {
        int o0 = wv * 16;
        v8f acc = {0.f, 0.f, 0.f, 0.f, 0.f, 0.f, 0.f, 0.f};
        for (int k0 = 0; k0 < D_; k0 += 4) {
            int k = k0 + 2 * half;
            v2f a, bb;
            a[0]  = xsum[l16][k];
            a[1]  = xsum[l16][k + 1];
            bb[0] = Wk[(o0 + l16) * D_ + k];        // B[k][n] = Wk[o][d]
            bb[1] = Wk[(o0 + l16) * D_ + k + 1];
            acc = __builtin_amdgcn_wmma_f32_16x16x4_f32(
                false, a, false, bb, (short)0, acc, false, false);
        }
        float bias = (float)N_ * bk[o0 + l16];      // same o for all 8 rows held here
        for (int r = 0; r < 8; ++r) {
            int m = r + 8 * half;
            ksum[m][o0 + l16] = acc[r] + bias;
        }
    }
    __syncthreads();

    // --- stage C: v = scale * (ksum * Wq) ---
    {
        int d0 = wv * 16;
        v8f acc = {0.f, 0.f, 0.f, 0.f, 0.f, 0.f, 0.f, 0.f};
        for (int k0 = 0; k0 < O_; k0 += 4) {
            int k = k0 + 2 * half;
            v2f a, bb;
            a[0]  = ksum[l16][k];
            a[1]  = ksum[l16][k + 1];
            bb[0] = Wq[k * D_ + d0 + l16];          // B[k][n] = Wq[o][d]
            bb[1] = Wq[(k + 1) * D_ + d0 + l16];
            acc = __builtin_amdgcn_wmma_f32_16x16x4_f32(
                false, a, false, bb, (short)0, acc, false, false);
        }
        for (int r = 0; r < 8; ++r) {
            int m = r + 8 * half;
            ws[OFF_V + (unsigned)(m * D_ + d0 + l16)] = SCALE_ * acc[r];
        }
    }

    // --- c[b] ---
    if (tid < B_) {
        float c = 0.f;
        for (int o = 0; o < O_; ++o) c += bq[o] * ksum[tid][o];
        ws[OFF_C + (unsigned)tid] = SCALE_ * c;
    }
}

// ---------------------------------------------------------------------------
// Pass 3 (fused): per row n:
//   s[b,n] = dot(x[b,n,:], v[b,:]) + c[b]      (scale pre-folded into v,c)
// and online-softmax weighted accumulation of x into per-block partials:
//   acc = acc*exp(m_old-m_new) + exp(s-m_new)*x_row
// Emits per block: aggp[b][blk][d], pmax[b][blk], psum[b][blk], plus s.
// ---------------------------------------------------------------------------
__global__ __launch_bounds__(256)
void k3_fused(const float* __restrict__ x, float* __restrict__ ws) {
    __shared__ float vsh[D_];
    __shared__ float accsh[8][D_];
    __shared__ float msh[8];
    __shared__ float lsh[8];
    int blk = blockIdx.x;                // b*NB3 + sl
    int b = blk / NB3, sl = blk % NB3;
    int tid = threadIdx.x, lane = tid & 31, wv = tid >> 5;

    vsh[tid] = ws[OFF_V + (unsigned)(b * D_ + tid)];
    __syncthreads();

    float cb = ws[OFF_C + (unsigned)b];
    int base = sl * ROWS3;
    const float* xb = x + (size_t)b * N_ * D_;
    float* sb = ws + OFF_S + (size_t)b * N_;

    const float* vp = vsh + lane * 8;
    float4 v0 = ((const float4*)vp)[0];
    float4 v1 = ((const float4*)vp)[1];

    float m_run = -3.402823466e38f;
    float l_run = 0.f;
    float4 a0 = {0.f, 0.f, 0.f, 0.f};
    float4 a1 = {0.f, 0.f, 0.f, 0.f};

    for (int r = wv; r < ROWS3; r += 8) {
        int row = base + r;
        const float* xr = xb + (size_t)row * D_ + lane * 8;
        if (r + 8 < ROWS3) __builtin_prefetch(xr + (size_t)8 * D_, 0, 0);
        float4 x0 = ((const float4*)xr)[0];
        float4 x1 = ((const float4*)xr)[1];
        float p = x0.x * v0.x + x0.y * v0.y + x0.z * v0.z + x0.w * v0.w
                + x1.x * v1.x + x1.y * v1.y + x1.z * v1.z + x1.w * v1.w;
        for (int off = 16; off > 0; off >>= 1) p += __shfl_xor(p, off, 32);
        float sval = p + cb;                 // uniform across the wave
        if (lane == 0) sb[row] = sval;

        float newm = fmaxf(m_run, sval);
        float sc = __expf(m_run - newm);     // rescale old accumulators
        float w  = __expf(sval - newm);      // this row's weight (unnormalized)
        l_run = l_run * sc + w;
        a0.x = a0.x * sc + w * x0.x;  a0.y = a0.y * sc + w * x0.y;
        a0.z = a0.z * sc + w * x0.z;  a0.w = a0.w * sc + w * x0.w;
        a1.x = a1.x * sc + w * x1.x;  a1.y = a1.y * sc + w * x1.y;
        a1.z = a1.z * sc + w * x1.z;  a1.w = a1.w * sc + w * x1.w;
        m_run = newm;
    }

    float* ap = &accsh[wv][lane * 8];
    ((float4*)ap)[0] = a0;
    ((float4*)ap)[1] = a1;
    if (lane == 0) { msh[wv] = m_run; lsh[wv] = l_run; }
    __syncthreads();

    // block combine (deterministic fixed order over the 8 waves)
    float mb = msh[0];
    for (int i = 1; i < 8; ++i) mb = fmaxf(mb, msh[i]);
    float agg = 0.f;
    for (int w2 = 0; w2 < 8; ++w2)
        agg += __expf(msh[w2] - mb) * accsh[w2][tid];
    ws[OFF_AGGP + (unsigned)((b * NB3 + sl) * D_ + tid)] = agg;
    if (tid == 0) {
        float lb = 0.f;
        for (int w2 = 0; w2 < 8; ++w2) lb += __expf(msh[w2] - mb) * lsh[w2];
        ws[OFF_PMAX + (unsigned)(b * NB3 + sl)] = mb;
        ws[OFF_PSUM + (unsigned)(b * NB3 + sl)] = lb;
    }
}

// ---------------------------------------------------------------------------
// Pass 4: combine NB3 block partials per batch with the global max:
//   m[b], linv[b] = 1/sum, and aggregated_feature -> d_out[0:4096]
// ---------------------------------------------------------------------------
__global__ __launch_bounds__(256)
void k4_finalize(float* __restrict__ ws, float* __restrict__ out) {
    int b = blockIdx.x, tid = threadIdx.x;   // tid == d

    float m = -3.402823466e38f;
    for (int k = 0; k < NB3; ++k)
        m = fmaxf(m, ws[OFF_PMAX + (unsigned)(b * NB3 + k)]);
    float l = 0.f;
    for (int k = 0; k < NB3; ++k)
        l += __expf(ws[OFF_PMAX + (unsigned)(b * NB3 + k)] - m)
           * ws[OFF_PSUM + (unsigned)(b * NB3 + k)];
    float linv = 1.0f / l;

    float agg = 0.f;
    for (int k = 0; k < NB3; ++k)
        agg += __expf(ws[OFF_PMAX + (unsigned)(b * NB3 + k)] - m)
             * ws[OFF_AGGP + (unsigned)((b * NB3 + k) * D_ + tid)];
    out[b * D_ + tid] = agg * linv;

    if (tid == 0) {
        ws[OFF_M + (unsigned)b]    = m;
        ws[OFF_LINV + (unsigned)b] = linv;
    }
}

// ---------------------------------------------------------------------------
// Pass 5: average_customer_weights[n] = mean_b w[b,n]   (d_out[4096:20480])
// ---------------------------------------------------------------------------
__global__ __launch_bounds__(256)
void k7_avg_weights(const float* __restrict__ ws, float* __restrict__ out) {
    int n = blockIdx.x * 256 + threadIdx.x;   // 0..16383
    float acc = 0.f;
    for (int b = 0; b < B_; ++b) {
        float m    = ws[OFF_M + (unsigned)b];
        float linv = ws[OFF_LINV + (unsigned)b];
        acc += __expf(ws[OFF_S + (size_t)b * N_ + n] - m) * linv;
    }
    out[B_ * D_ + n] = acc * (1.0f / (float)B_);
}

// ---------------------------------------------------------------------------
extern "C" void kernel_launch(void* const* d_in, const int* in_sizes, int n_in,
                              void* d_out, int out_size, void* d_ws, size_t ws_size,
                              hipStream_t stream) {
    const float* x  = (const float*)d_in[0];
    const float* Wq = (const float*)d_in[1];
    const float* bq = (const float*)d_in[2];
    const float* Wk = (const float*)d_in[3];
    const float* bk = (const float*)d_in[4];
    float* out = (float*)d_out;
    float* ws  = (float*)d_ws;

    k1_partial_sum  <<<B_ * SLICES1, 256, 0, stream>>>(x, ws);
    k2_small_gemms  <<<1, 512, 0, stream>>>(Wq, bq, Wk, bk, ws);
    k3_fused        <<<B_ * NB3, 256, 0, stream>>>(x, ws);
    k4_finalize     <<<B_, 256, 0, stream>>>(ws, out);
    k7_avg_weights  <<<N_ / 256, 256, 0, stream>>>(ws, out);
}